// LengthRegulator_33088428049214
// MI455X (gfx1250) — compile-verified
//
#include <hip/hip_runtime.h>
#include <math.h>
#include <stdint.h>

#define B_   32
#define TX_  512
#define TY_  2048
#define C_   384
#define MU_ELEMS   ((size_t)B_ * TY_ * C_)    // 25165824
#define ATTN_ELEMS ((size_t)B_ * TX_ * TY_)   // 33554432

typedef __attribute__((ext_vector_type(2))) float v2f;
typedef __attribute__((ext_vector_type(8))) float v8f;

// ---------------------------------------------------------------------------
// Kernel 1: per-batch cumsum of durations, duration-loss partial, ty->tx map.
// One block per batch, 512 threads (16 waves).
// ---------------------------------------------------------------------------
__global__ void k_scan(const float* __restrict__ dur,
                       const int*   __restrict__ xlen,
                       const int*   __restrict__ ylen,
                       const float* __restrict__ logw,
                       int*         __restrict__ map,
                       float*       __restrict__ num) {
    __shared__ int   s[TX_];
    __shared__ float red[TX_];
    const int b = blockIdx.x;
    const int t = threadIdx.x;
    const int xl = xlen[b];
    const int yl = ylen[b];

    int d = (t < xl) ? (int)dur[b * TX_ + t] : 0;
    s[t] = d;
    __syncthreads();
    // Hillis-Steele inclusive scan
    for (int off = 1; off < TX_; off <<= 1) {
        int v = (t >= off) ? s[t - off] : 0;
        __syncthreads();
        s[t] += v;
        __syncthreads();
    }
    const int chi = s[t];
    const int clo = (t > 0) ? s[t - 1] : 0;
    const int lo  = min(clo, yl);
    const int hi  = min(chi, yl);

    // logw_gt = (t < xl) ? log(1e-8 + rowsum) : 0   (rowsum = hi - lo)
    float lgt = 0.0f;
    if (t < xl) lgt = logf(1e-8f + (float)(hi - lo));
    const float diff = logw[b * TX_ + t] - lgt;
    red[t] = diff * diff;
    __syncthreads();
    for (int off = TX_ / 2; off > 0; off >>= 1) {
        if (t < off) red[t] += red[t + off];
        __syncthreads();
    }
    if (t == 0) atomicAdd(num, red[0]);

    // scatter inverse map (map pre-initialized to -1 via memset)
    if (t < xl) {
        int* mb = map + b * TY_;
        for (int ty = lo; ty < hi; ++ty) mb[ty] = t;
    }
}

// ---------------------------------------------------------------------------
// Kernel 2: materialize attn[b,tx,ty] = (map[b,ty] == tx). Coalesced b32.
// ---------------------------------------------------------------------------
__global__ void k_attn(const int* __restrict__ map,
                       float*     __restrict__ attn) {
    size_t idx = (size_t)blockIdx.x * blockDim.x + threadIdx.x;
    int ty = (int)(idx % TY_);
    int tx = (int)((idx / TY_) % TX_);
    int b  = (int)(idx / ((size_t)TY_ * TX_));
    int m  = map[b * TY_ + ty];
    attn[idx] = (m == tx) ? 1.0f : 0.0f;
}

// ---------------------------------------------------------------------------
// Kernel 3: mu_y[b,ty,c] = sum_tx attn[b,tx,ty] * x[b,tx,c] via f32 WMMA.
// One wave per 16x16 (ty x c) output tile. Because durations >= 1, the 16 ty
// rows of a tile map to a contiguous tx span of width <= 16, so the K loop is
// <= 5 chunks of K=4 (V_WMMA_F32_16X16X4_F32), exact f32 math.
// Out-of-span K values have A==0, so B only needs a *safe* (clamped) address,
// not a zero value -> unconditional coalesced loads, no exec-divergence.
// Block = 256 threads = 8 waves = 8 tiles.
// ---------------------------------------------------------------------------
__global__ void k_mu(const float* __restrict__ x,
                     const int*   __restrict__ map,
                     float*       __restrict__ mu) {
    const int cT  = C_ / 16;    // 24
    const int tyT = TY_ / 16;   // 128
    const int wave = threadIdx.x >> 5;
    const int lane = threadIdx.x & 31;
    const int tile = blockIdx.x * (blockDim.x >> 5) + wave;

    const int b   = tile / (tyT * cT);
    const int rem = tile % (tyT * cT);
    const int tyB = (rem / cT) * 16;
    const int cB  = (rem % cT) * 16;

    const int l16 = lane >> 4;   // 0 or 1
    const int r   = lane & 15;

    // A-matrix row for lane: M = r  -> its source tx
    const int m = map[b * TY_ + tyB + r];

    // uniform min/max of valid tx over the 16 rows (wave reduction)
    int mmin = (m < 0) ? 0x7fffffff : m;
    int mmax = m;
    #pragma unroll
    for (int off = 16; off >= 1; off >>= 1) {
        mmin = min(mmin, __shfl_xor(mmin, off, 32));
        mmax = max(mmax, __shfl_xor(mmax, off, 32));
    }

    v8f acc = {};
    if (mmax >= 0) {                       // uniform across wave -> EXEC full
        const float* xb = x + (size_t)b * TX_ * C_ + cB + r;
        const int k0 = mmin & ~3;
        for (int kb = k0; kb <= mmax; kb += 4) {
            // A 16x4: lane holds M=r, K = v + 2*(lane/16)
            // B 4x16: lane holds N=r, K = v + 2*(lane/16)
            const int k  = kb + 2 * l16;
            const int ka = min(k,     TX_ - 1);   // safe addr; A==0 past mmax
            const int kb2 = min(k + 1, TX_ - 1);
            v2f A, Bv;
            A.x  = (m == k)     ? 1.0f : 0.0f;
            A.y  = (m == k + 1) ? 1.0f : 0.0f;
            Bv.x = xb[(size_t)ka  * C_];
            Bv.y = xb[(size_t)kb2 * C_];
            acc = __builtin_amdgcn_wmma_f32_16x16x4_f32(
                /*neg_a=*/false, A, /*neg_b=*/false, Bv,
                /*c_mod=*/(short)0, acc, /*reuse_a=*/false, /*reuse_b=*/false);
        }
    }

    // D 16x16 f32 layout: lane holds column N=r; VGPR v holds row M = v + 8*(lane/16)
    float* mb = mu + (size_t)b * TY_ * C_ + cB + r;
    #pragma unroll
    for (int v = 0; v < 8; ++v) {
        const int ty = tyB + v + 8 * l16;
        mb[(size_t)ty * C_] = acc[v];
    }
}

// ---------------------------------------------------------------------------
// Kernel 4: dur_loss = num / sum(x_lengths)
// ---------------------------------------------------------------------------
__global__ void k_fin(const int*   __restrict__ xlen,
                      const float* __restrict__ num,
                      float*       __restrict__ loss) {
    int s = 0;
    for (int b = 0; b < B_; ++b) s += xlen[b];
    *loss = *num / (float)s;
}

extern "C" void kernel_launch(void* const* d_in, const int* in_sizes, int n_in,
                              void* d_out, int out_size, void* d_ws, size_t ws_size,
                              hipStream_t stream) {
    const float* x    = (const float*)d_in[0];
    const int*   xlen = (const int*)  d_in[1];
    const int*   ylen = (const int*)  d_in[4];
    const float* logw = (const float*)d_in[6];
    const float* dur  = (const float*)d_in[7];

    float* out  = (float*)d_out;
    float* mu   = out;                     // (B, TY, C)
    float* loss = out + MU_ELEMS;          // scalar
    float* attn = out + MU_ELEMS + 1;      // (B, TX, TY) -- 4B aligned only

    float* num = (float*)d_ws;
    int*   map = (int*)((char*)d_ws + 64);

    hipMemsetAsync(num, 0, sizeof(float), stream);
    hipMemsetAsync(map, 0xFF, (size_t)B_ * TY_ * sizeof(int), stream);  // -1

    k_scan<<<B_, TX_, 0, stream>>>(dur, xlen, ylen, logw, map, num);

    k_attn<<<(unsigned)(ATTN_ELEMS / 256), 256, 0, stream>>>(map, attn);

    const int tiles = B_ * (TY_ / 16) * (C_ / 16);   // 98304
    k_mu<<<tiles / 8, 256, 0, stream>>>(x, map, mu);

    k_fin<<<1, 1, 0, stream>>>(xlen, num, loss);
}